// Detector_51299089383657
// MI455X (gfx1250) — compile-verified
//
#include <hip/hip_runtime.h>
#include <math.h>

#define NANCH 8732
#define NCLS 91
#define NCLS1 90
#define TOPK 400
#define DETS 200
#define BATCH 8
#define IMGSZ 300.0f
#define SCORE_TH 0.01f
#define NMS_TH 0.45f
#define XFORM_CLIP 4.135166556742356f /* log(1000/16) */

typedef float v2f __attribute__((ext_vector_type(2)));
typedef float v8f __attribute__((ext_vector_type(8)));
typedef int av4i __attribute__((vector_size(16)));  // async-to-LDS b128 payload

// ---------------------------------------------------------------------------
// Kernel 1: per-anchor softmax statistics (max + 1/sum(exp)).
// One wave owns 16 anchors; lane l and lane l^16 split the 91 classes of
// anchor (l&15). All 46 per-lane loads are issued unconditionally up front
// (invalid lanes read anchor 0; the class-91 pad slot reads class 0 and is
// zeroed with a select) so the memory pipe stays full — no EXEC-mask
// branching in the hot loop. Sum of exps runs on the matrix pipe via 23
// chained V_WMMA_F32_16X16X4_F32 with B == ones (row-sum).
// ---------------------------------------------------------------------------
__global__ __launch_bounds__(256)
void softmax_stats_kernel(const float* __restrict__ logits,
                          float* __restrict__ maxv,
                          float* __restrict__ rden) {
  const int b = blockIdx.y;
  const int wave = blockIdx.x * (blockDim.x >> 5) + (threadIdx.x >> 5);
  const int lane = threadIdx.x & 31;
  const int half = lane >> 4;
  const int anchor0 = wave * 16 + (lane & 15);
  const bool valid = anchor0 < NANCH;
  const int anchor = valid ? anchor0 : 0;  // clamp: loads always in-bounds
  const float* lg = logits + ((size_t)b * NANCH + anchor) * NCLS;

  // one streaming pass: preload this lane's 46 class logits
  float x[46];
#pragma unroll
  for (int t = 0; t < 23; ++t) {
    const int k0 = 4 * t + 2 * half;
    x[2 * t] = lg[k0];                                    // k0 <= 90, in-bounds
    x[2 * t + 1] = lg[(k0 + 1 < NCLS) ? (k0 + 1) : 0];    // pad slot -> class 0
  }

  // per-anchor max (duplicate class-0 read is harmless: class 0 is in the set)
  float m = -3.0e38f;
#pragma unroll
  for (int i = 0; i < 46; ++i) m = fmaxf(m, x[i]);
  m = fmaxf(m, __shfl_xor(m, 16, 32));

  float denom;
#if defined(__gfx1250__) && __has_builtin(__builtin_amdgcn_wmma_f32_16x16x4_f32)
  // A layout (16x4 f32): lane l holds M=l%16, K = 2*(l/16) + {vgpr0,vgpr1}.
  v8f acc = {0.f, 0.f, 0.f, 0.f, 0.f, 0.f, 0.f, 0.f};
  v2f bones;
  bones.x = 1.0f;
  bones.y = 1.0f;
#pragma unroll
  for (int t = 0; t < 23; ++t) {
    float e0 = __expf(x[2 * t] - m);
    float e1 = __expf(x[2 * t + 1] - m);
    if (t == 22) e1 = half ? 0.0f : e1;  // mask padded class-91 slot (cndmask)
    v2f a;
    a.x = e0;
    a.y = e1;
    acc = __builtin_amdgcn_wmma_f32_16x16x4_f32(false, a, false, bones,
                                                (short)0, acc, false, false);
  }
  // D layout: vgpr r, lanes<16 -> M=r; lanes>=16 -> M=r+8. Every column equal.
  // Lane l wants M = l&15: select component l&7, take partner (shfl_xor 16)
  // when bit3(l) ^ bit4(l) == 1.
  float s01 = (lane & 1) ? acc[1] : acc[0];
  float s23 = (lane & 1) ? acc[3] : acc[2];
  float s45 = (lane & 1) ? acc[5] : acc[4];
  float s67 = (lane & 1) ? acc[7] : acc[6];
  float s03 = (lane & 2) ? s23 : s01;
  float s47 = (lane & 2) ? s67 : s45;
  float comp = (lane & 4) ? s47 : s03;
  float part = __shfl_xor(comp, 16, 32);
  bool own = (((lane >> 3) & 1) ^ ((lane >> 4) & 1)) == 0;
  denom = own ? comp : part;
#else
  float s = 0.0f;
#pragma unroll
  for (int t = 0; t < 23; ++t) {
    float e0 = __expf(x[2 * t] - m);
    float e1 = __expf(x[2 * t + 1] - m);
    if (t == 22) e1 = half ? 0.0f : e1;
    s += e0 + e1;
  }
  denom = s + __shfl_xor(s, 16, 32);
#endif

  if (valid && half == 0) {
    maxv[(size_t)b * NANCH + anchor0] = m;
    rden[(size_t)b * NANCH + anchor0] = 1.0f / denom;
  }
}

// ---------------------------------------------------------------------------
// Kernel 2: decode + clip boxes (torchvision BoxCoder, weights 10,10,5,5).
// ---------------------------------------------------------------------------
__global__ __launch_bounds__(256)
void decode_kernel(const float* __restrict__ reg,
                   const float* __restrict__ anchors,
                   float* __restrict__ boxes) {
  int t = blockIdx.x * blockDim.x + threadIdx.x;
  if (t >= BATCH * NANCH) return;
  int n = t % NANCH;
  float4 r = ((const float4*)reg)[t];
  float4 a = ((const float4*)anchors)[n];
  float aw = a.z - a.x, ah = a.w - a.y;
  float ax = a.x + 0.5f * aw, ay = a.y + 0.5f * ah;
  float dx = r.x * 0.1f, dy = r.y * 0.1f;
  float dw = fminf(r.z * 0.2f, XFORM_CLIP);
  float dh = fminf(r.w * 0.2f, XFORM_CLIP);
  float px = dx * aw + ax, py = dy * ah + ay;
  float pw = __expf(dw) * aw, ph = __expf(dh) * ah;
  float4 o;
  o.x = fminf(fmaxf(px - 0.5f * pw, 0.f), IMGSZ);
  o.y = fminf(fmaxf(py - 0.5f * ph, 0.f), IMGSZ);
  o.z = fminf(fmaxf(px + 0.5f * pw, 0.f), IMGSZ);
  o.w = fminf(fmaxf(py + 0.5f * ph, 0.f), IMGSZ);
  ((float4*)boxes)[t] = o;
}

// ---------------------------------------------------------------------------
// Kernel 3: per (image, class): exact top-400 via 4-pass radix select on the
// float bit patterns (scores >= 0 so uint order == float order), bitonic sort
// 512 (key desc, index asc), greedy NMS on LDS-staged boxes.
// ---------------------------------------------------------------------------
__global__ __launch_bounds__(256)
void topk_nms_kernel(const float* __restrict__ logits,
                     const float* __restrict__ maxv,
                     const float* __restrict__ rden,
                     const float* __restrict__ boxes,
                     float* __restrict__ candS,
                     float* __restrict__ candB) {
  const int c = blockIdx.x;  // class c+1 (background dropped)
  const int b = blockIdx.y;
  const int tid = threadIdx.x;

  __shared__ float sscore[NANCH];
  __shared__ unsigned hist[256];
  __shared__ unsigned skey[512];
  __shared__ unsigned sidx[512];
  __shared__ __attribute__((aligned(16))) float sbx[512 * 4];
  __shared__ float sarea[512];
  __shared__ unsigned char keep[512];
  __shared__ unsigned sh_prefix, sh_mask, sh_above, sh_hi, sh_tie;

  // scores for this class into LDS (logits are L2-resident: 25MB << 192MB L2)
  const float* lg = logits + (size_t)b * NANCH * NCLS + (c + 1);
  const float* mv = maxv + (size_t)b * NANCH;
  const float* rd = rden + (size_t)b * NANCH;
  for (int n = tid; n < NANCH; n += 256)
    sscore[n] = __expf(lg[(size_t)n * NCLS] - mv[n]) * rd[n];
  if (tid == 0) { sh_prefix = 0u; sh_mask = 0u; sh_above = 0u; }
  __syncthreads();

  // radix select the exact bit pattern of the 400th-largest score
  for (int pass = 0; pass < 4; ++pass) {
    int shift = 24 - 8 * pass;
    hist[tid] = 0u;
    __syncthreads();
    unsigned pfx = sh_prefix, msk = sh_mask;
    for (int n = tid; n < NANCH; n += 256) {
      unsigned bits = __float_as_uint(sscore[n]);
      if ((bits & msk) == pfx) atomicAdd(&hist[(bits >> shift) & 0xFFu], 1u);
    }
    __syncthreads();
    if (tid == 0) {
      unsigned need = TOPK - sh_above, cum = 0u;
      int bsel = 0;
      for (int v = 255; v >= 0; --v) {
        if (cum + hist[v] >= need) { bsel = v; break; }
        cum += hist[v];
      }
      sh_above += cum;
      sh_prefix |= ((unsigned)bsel) << shift;
      sh_mask |= 0xFFu << shift;
    }
    __syncthreads();
  }
  const unsigned T = sh_prefix;
  const unsigned nAbove = sh_above;  // #(bits > T), < 400 by construction

  for (int i = tid; i < 512; i += 256) { skey[i] = 0u; sidx[i] = 0u; }
  if (tid == 0) { sh_hi = 0u; sh_tie = 0u; }
  __syncthreads();
  for (int n = tid; n < NANCH; n += 256) {
    unsigned bits = __float_as_uint(sscore[n]);
    if (bits > T) {
      unsigned p = atomicAdd(&sh_hi, 1u);
      skey[p] = bits; sidx[p] = (unsigned)n;
    } else if (bits == T) {
      unsigned p = nAbove + atomicAdd(&sh_tie, 1u);
      if (p < TOPK) { skey[p] = bits; sidx[p] = (unsigned)n; }
    }
  }
  __syncthreads();

  // bitonic sort 512: key descending, index ascending on ties (deterministic)
  for (unsigned k = 2; k <= 512; k <<= 1) {
    for (unsigned j = k >> 1; j > 0; j >>= 1) {
      __syncthreads();
      for (unsigned i = tid; i < 512; i += 256) {
        unsigned x = i ^ j;
        if (x > i) {
          unsigned ki = skey[i], kx = skey[x], vi = sidx[i], vx = sidx[x];
          bool iBefore = (ki > kx) || (ki == kx && vi < vx);
          bool desc = ((i & k) == 0);
          if (desc ? !iBefore : iBefore) {
            skey[i] = kx; skey[x] = ki;
            sidx[i] = vx; sidx[x] = vi;
          }
        }
      }
    }
  }
  __syncthreads();

  // gather candidate boxes into LDS (async DMA path on gfx1250)
  for (int i = tid; i < TOPK; i += 256) {
    unsigned a = sidx[i];
    const float* src = boxes + ((size_t)b * NANCH + a) * 4;
#if defined(__gfx1250__) && __has_builtin(__builtin_amdgcn_global_load_async_to_lds_b128)
    __builtin_amdgcn_global_load_async_to_lds_b128(
        (__attribute__((address_space(1))) av4i*)src,
        (__attribute__((address_space(3))) av4i*)&sbx[i * 4], 0, 0);
#else
    float4 bb = *(const float4*)src;
    sbx[i * 4 + 0] = bb.x; sbx[i * 4 + 1] = bb.y;
    sbx[i * 4 + 2] = bb.z; sbx[i * 4 + 3] = bb.w;
#endif
    keep[i] = 1;
  }
#if defined(__gfx1250__) && __has_builtin(__builtin_amdgcn_s_wait_asynccnt)
  __builtin_amdgcn_s_wait_asynccnt(0);
#endif
  __syncthreads();
  for (int i = tid; i < TOPK; i += 256)
    sarea[i] = (sbx[i * 4 + 2] - sbx[i * 4 + 0]) * (sbx[i * 4 + 3] - sbx[i * 4 + 1]);
  __syncthreads();

  // greedy NMS over sorted candidates
  for (int i = 0; i < TOPK - 1; ++i) {
    __syncthreads();
    if (!keep[i]) continue;  // uniform branch (LDS scalar)
    float ix0 = sbx[i * 4 + 0], iy0 = sbx[i * 4 + 1];
    float ix1 = sbx[i * 4 + 2], iy1 = sbx[i * 4 + 3];
    float ia = sarea[i];
    for (int j = i + 1 + tid; j < TOPK; j += 256) {
      if (!keep[j]) continue;
      float ltx = fmaxf(ix0, sbx[j * 4 + 0]);
      float lty = fmaxf(iy0, sbx[j * 4 + 1]);
      float rbx = fminf(ix1, sbx[j * 4 + 2]);
      float rby = fminf(iy1, sbx[j * 4 + 3]);
      float w = fmaxf(rbx - ltx, 0.f), h = fmaxf(rby - lty, 0.f);
      float inter = w * h;
      float iou = inter / (ia + sarea[j] - inter + 1e-8f);
      if (iou > NMS_TH) keep[j] = 0;
    }
  }
  __syncthreads();

  size_t base = ((size_t)b * NCLS1 + c) * TOPK;
  for (int i = tid; i < TOPK; i += 256) {
    float v = __uint_as_float(skey[i]);
    candS[base + i] = (keep[i] && v > SCORE_TH) ? v : 0.0f;
    candB[(base + i) * 4 + 0] = sbx[i * 4 + 0];
    candB[(base + i) * 4 + 1] = sbx[i * 4 + 1];
    candB[(base + i) * 4 + 2] = sbx[i * 4 + 2];
    candB[(base + i) * 4 + 3] = sbx[i * 4 + 3];
  }
}

// ---------------------------------------------------------------------------
// Kernel 4: per-image top-200 over 90*400 candidates -> [200, 6] rows.
// ---------------------------------------------------------------------------
__global__ __launch_bounds__(256)
void final_topk_kernel(const float* __restrict__ candS,
                       const float* __restrict__ candB,
                       float* __restrict__ out) {
  const int b = blockIdx.x;
  const int tid = threadIdx.x;
  const int M = NCLS1 * TOPK;  // 36000 (L2-resident)
  const float* sc = candS + (size_t)b * M;

  __shared__ unsigned hist[256];
  __shared__ unsigned skey[256];
  __shared__ unsigned sidx[256];
  __shared__ unsigned sh_prefix, sh_mask, sh_above, sh_hi, sh_tie;

  if (tid == 0) { sh_prefix = 0u; sh_mask = 0u; sh_above = 0u; }
  __syncthreads();
  for (int pass = 0; pass < 4; ++pass) {
    int shift = 24 - 8 * pass;
    hist[tid] = 0u;
    __syncthreads();
    unsigned pfx = sh_prefix, msk = sh_mask;
    for (int n = tid; n < M; n += 256) {
      unsigned bits = __float_as_uint(sc[n]);
      if ((bits & msk) == pfx) atomicAdd(&hist[(bits >> shift) & 0xFFu], 1u);
    }
    __syncthreads();
    if (tid == 0) {
      unsigned need = DETS - sh_above, cum = 0u;
      int bsel = 0;
      for (int v = 255; v >= 0; --v) {
        if (cum + hist[v] >= need) { bsel = v; break; }
        cum += hist[v];
      }
      sh_above += cum;
      sh_prefix |= ((unsigned)bsel) << shift;
      sh_mask |= 0xFFu << shift;
    }
    __syncthreads();
  }
  const unsigned T = sh_prefix;
  const unsigned nAbove = sh_above;

  skey[tid] = 0u; sidx[tid] = 0u;
  if (tid == 0) { sh_hi = 0u; sh_tie = 0u; }
  __syncthreads();
  for (int n = tid; n < M; n += 256) {
    unsigned bits = __float_as_uint(sc[n]);
    if (bits > T) {
      unsigned p = atomicAdd(&sh_hi, 1u);
      skey[p] = bits; sidx[p] = (unsigned)n;
    } else if (bits == T) {
      unsigned p = nAbove + atomicAdd(&sh_tie, 1u);
      if (p < DETS) { skey[p] = bits; sidx[p] = (unsigned)n; }
    }
  }
  __syncthreads();

  for (unsigned k = 2; k <= 256; k <<= 1) {
    for (unsigned j = k >> 1; j > 0; j >>= 1) {
      __syncthreads();
      unsigned i = tid, x = i ^ j;
      if (x > i) {
        unsigned ki = skey[i], kx = skey[x], vi = sidx[i], vx = sidx[x];
        bool iBefore = (ki > kx) || (ki == kx && vi < vx);
        bool desc = ((i & k) == 0);
        if (desc ? !iBefore : iBefore) {
          skey[i] = kx; skey[x] = ki;
          sidx[i] = vx; sidx[x] = vi;
        }
      }
    }
  }
  __syncthreads();

  for (int i = tid; i < DETS; i += 256) {
    unsigned ci = sidx[i];
    const float* bb = candB + ((size_t)b * M + ci) * 4;
    float* row = out + ((size_t)b * DETS + i) * 6;
    row[0] = bb[0]; row[1] = bb[1]; row[2] = bb[2]; row[3] = bb[3];
    row[4] = __uint_as_float(skey[i]);
    row[5] = (float)(ci / TOPK + 1);  // label
  }
}

// ---------------------------------------------------------------------------
extern "C" void kernel_launch(void* const* d_in, const int* in_sizes, int n_in,
                              void* d_out, int out_size, void* d_ws,
                              size_t ws_size, hipStream_t stream) {
  (void)in_sizes; (void)n_in; (void)out_size; (void)ws_size;
  const float* reg = (const float*)d_in[0];      // [8, 8732, 4]
  const float* logits = (const float*)d_in[1];   // [8, 8732, 91]
  const float* anchors = (const float*)d_in[2];  // [8732, 4]
  float* out = (float*)d_out;                    // [8, 200, 6]

  char* ws = (char*)d_ws;
  size_t off = 0;
  auto wsAlloc = [&](size_t bytes) -> void* {
    void* p = ws + off;
    off += (bytes + 255) & ~(size_t)255;
    return p;
  };
  float* boxes = (float*)wsAlloc(sizeof(float) * BATCH * NANCH * 4);  // 1.1 MB
  float* maxv  = (float*)wsAlloc(sizeof(float) * BATCH * NANCH);      // 0.28 MB
  float* rden  = (float*)wsAlloc(sizeof(float) * BATCH * NANCH);      // 0.28 MB
  float* candS = (float*)wsAlloc(sizeof(float) * BATCH * NCLS1 * TOPK);     // 1.15 MB
  float* candB = (float*)wsAlloc(sizeof(float) * BATCH * NCLS1 * TOPK * 4); // 4.6 MB

  const int wavesPerBatch = (NANCH + 15) / 16;               // 546
  dim3 g1((wavesPerBatch + 7) / 8, BATCH);
  softmax_stats_kernel<<<g1, 256, 0, stream>>>(logits, maxv, rden);

  decode_kernel<<<(BATCH * NANCH + 255) / 256, 256, 0, stream>>>(reg, anchors, boxes);

  dim3 g3(NCLS1, BATCH);  // 720 blocks
  topk_nms_kernel<<<g3, 256, 0, stream>>>(logits, maxv, rden, boxes, candS, candB);

  final_topk_kernel<<<BATCH, 256, 0, stream>>>(candS, candB, out);
}